// GCNNLayer_71536975282326
// MI455X (gfx1250) — compile-verified
//
#include <hip/hip_runtime.h>
#include <hip/hip_bf16.h>

typedef __attribute__((ext_vector_type(2))) float v2f;
typedef __attribute__((ext_vector_type(8))) float v8f;

namespace {
constexpr int N = 150;    // tokens
constexpr int D = 1024;   // emb dim
constexpr int R = 92;     // dep types
constexpr int NG = 4;     // r-groups (parallelism for the W_dep stream)
constexpr int RG = R / NG;        // 23 dep types per group
constexpr int KC  = 64;   // K-chunk staged in LDS
constexpr int KCP = 66;   // padded stride (bank-conflict avoidance, keeps 8B align)
constexpr int ROWS = 160; // 150 padded to 10 tiles of 16
constexpr int CNT_PAD = 16384;    // floats reserved for counts in ws
}

// ---------------- counts scatter ----------------
__global__ void zero_f32_kernel(float* __restrict__ p, int n) {
    int i = blockIdx.x * blockDim.x + threadIdx.x;
    if (i < n) p[i] = 0.0f;
}

__global__ void count_kernel(const int* __restrict__ tok,
                             const int* __restrict__ typ,
                             float* __restrict__ counts, int E) {
    int i = blockIdx.x * blockDim.x + threadIdx.x;
    if (i < E) atomicAdd(&counts[tok[i] * R + typ[i]], 1.0f);
}

// ---------------- self term: out[n,e] = x[n]@W_pos[n] + b_pos[n,e] + counts[n]@b_dep[:,e]
// Per-token private 1024x1024 matrix -> pure streaming vec-mat (memory bound),
// coalesced along e. Writes un-ReLU'd partial into d_out.
__global__ __launch_bounds__(256) void self_kernel(
        const float* __restrict__ x,
        const float* __restrict__ Wp,
        const float* __restrict__ b_pos,
        const float* __restrict__ b_dep,
        const float* __restrict__ counts,
        float* __restrict__ out) {
    __shared__ float xs[D];
    const int n = blockIdx.y;
    const int e = blockIdx.x * 256 + threadIdx.x;

    for (int i = threadIdx.x; i < D; i += 256) xs[i] = x[n * D + i];
    __syncthreads();

    const float* W = Wp + (size_t)n * D * D + e;
    float acc = 0.0f;
#pragma unroll 8
    for (int d = 0; d < D; ++d) acc += xs[d] * W[(size_t)d * D];

    acc += b_pos[n * D + e];
    const float* cn = counts + n * R;
    for (int r = 0; r < R; ++r) acc += cn[r] * b_dep[r * D + e];

    out[n * D + e] = acc;
}

// ---------------- dep term via fp32 WMMA ----------------
// partial_g[n,e] = sum_{r in group g} counts[n,r] * (x[n] @ W_dep[r])[e]
// Grid: (64 column stripes) x (4 r-groups) = 256 workgroups so the 386 MB
// W_dep stream has enough WGPs to reach HBM roofline. Block = 320 threads =
// 10 waves; wave w owns M-tile rows [16w, 16w+16). Each W_dep element is read
// from HBM exactly once device-wide; x re-reads are L2-resident (600 KB).
__global__ __launch_bounds__(320) void dep_wmma_kernel(
        const float* __restrict__ x,
        const float* __restrict__ Wd,
        const float* __restrict__ counts,
        float* __restrict__ accbuf) {
    __shared__ float Xs[ROWS * KCP];  // 160 x 64 chunk of x (rows >= N zeroed)
    __shared__ float Bs[16 * KCP];    // W_dep stripe, transposed [col][k]

    const int tid   = threadIdx.x;
    const int lane  = tid & 31;
    const int w     = tid >> 5;       // 0..9  -> M tile
    const int mbase = w * 16;
    const int col0  = blockIdx.x * 16;
    const int g     = blockIdx.y;     // r-group
    const int r0    = g * RG;
    const int r1    = r0 + RG;
    const int mrow  = lane & 15;          // A/B lane row/col index
    const int rowA  = mbase + mrow;       // token row this lane feeds into A
    const int kHalf = (lane < 16) ? 0 : 2; // K pair selector per ISA layout

    v8f acc = {};

    for (int kc = 0; kc < D; kc += KC) {
        // Stage x chunk (coalesced along k); pad rows >= N with zeros.
        for (int idx = tid; idx < ROWS * KC; idx += 320) {
            const int row = idx >> 6;      // /KC
            const int k   = idx & (KC - 1);
            float v = 0.0f;
            if (row < N) v = x[row * D + kc + k];
            Xs[row * KCP + k] = v;
        }
        __syncthreads();

        for (int r = r0; r < r1; ++r) {
            // Stage B = W_dep[r][kc..kc+63][col0..col0+15], stored transposed.
            // Also prefetch the next r's stripe so its cachelines are in
            // flight during this r's 16-WMMA compute phase.
            for (int idx = tid; idx < KC * 16; idx += 320) {
                const int k = idx >> 4;
                const int c = idx & 15;
                const size_t off = (size_t)(kc + k) * D + (col0 + c);
                Bs[c * KCP + k] = Wd[(size_t)r * D * D + off];
                if (r + 1 < r1)
                    __builtin_prefetch(&Wd[(size_t)(r + 1) * D * D + off], 0, 1);
            }
            __syncthreads();

            const float cnt  = (rowA < N) ? counts[rowA * R + r] : 0.0f;
            const float* xrow = &Xs[rowA * KCP + kHalf];
            const float* brow = &Bs[mrow * KCP + kHalf];

#pragma unroll
            for (int k4 = 0; k4 < KC; k4 += 4) {
                v2f a, b;
                a.x = cnt * xrow[k4 + 0];
                a.y = cnt * xrow[k4 + 1];
                b.x = brow[k4 + 0];
                b.y = brow[k4 + 1];
                // D(16x16,f32) += A(16x4,f32) x B(4x16,f32)
                acc = __builtin_amdgcn_wmma_f32_16x16x4_f32(
                        false, a, false, b, (short)0, acc, false, false);
            }
            __syncthreads();
        }
    }

    // Write this group's partial tile (each element has a unique owner).
    float* accg = accbuf + (size_t)g * N * D;
#pragma unroll
    for (int v = 0; v < 8; ++v) {
        const int row = mbase + v + ((lane < 16) ? 0 : 8);
        const int col = col0 + mrow;
        if (row < N) accg[row * D + col] = acc[v];
    }
}

// ---------------- final fuse: out = relu(self + sum_g partial_g) ----------------
__global__ __launch_bounds__(256) void fuse_kernel(
        const float* __restrict__ accbuf, float* __restrict__ out) {
    const int i = blockIdx.x * 256 + threadIdx.x;  // grid covers N*D exactly
    float s = out[i];
#pragma unroll
    for (int g = 0; g < NG; ++g) s += accbuf[(size_t)g * N * D + i];
    out[i] = s > 0.0f ? s : 0.0f;
}

extern "C" void kernel_launch(void* const* d_in, const int* in_sizes, int n_in,
                              void* d_out, int out_size, void* d_ws, size_t ws_size,
                              hipStream_t stream) {
    const float* x      = (const float*)d_in[0];  // [N,D]
    const float* W_pos  = (const float*)d_in[1];  // [N,D,D]
    const float* b_pos  = (const float*)d_in[2];  // [N,D]
    const float* W_dep  = (const float*)d_in[3];  // [R,D,D]
    const float* b_dep  = (const float*)d_in[4];  // [R,D]
    const int*   e_tok  = (const int*)d_in[5];    // [E]
    const int*   e_typ  = (const int*)d_in[6];    // [E]
    float*       out    = (float*)d_out;          // [N,D]
    float*       counts = (float*)d_ws;           // [N,R] scratch
    float*       accbuf = counts + CNT_PAD;       // [NG,N,D] partials

    const int E = in_sizes[5];
    const int nc = N * R;

    zero_f32_kernel<<<(nc + 255) / 256, 256, 0, stream>>>(counts, nc);
    count_kernel<<<(E + 255) / 256, 256, 0, stream>>>(e_tok, e_typ, counts, E);
    self_kernel<<<dim3(D / 256, N), 256, 0, stream>>>(x, W_pos, b_pos, b_dep,
                                                      counts, out);
    dep_wmma_kernel<<<dim3(D / 16, NG), 320, 0, stream>>>(x, W_dep, counts,
                                                          accbuf);
    fuse_kernel<<<(N * D) / 256, 256, 0, stream>>>(accbuf, out);
}